// CTC_3667902071358
// MI455X (gfx1250) — compile-verified
//
#include <hip/hip_runtime.h>
#include <stdint.h>

// CTC loss forward on gfx1250 (MI455X).
//   B=32, T=1024, V=1024, LMAX=128, S=2*LMAX+1=257, BLANK=0
// Phase 1: per-(b,t) logsumexp over V (bandwidth bound, one wave32 per row).
// Phase 2: sequential CTC DP, one block per batch element; per-step gather of
//          ys_hat[b,t,ext[s]] is prefetched with CDNA5 async global->LDS loads
//          (ASYNCcnt double buffer) so global latency is off the serial path.
//          Loop runs only to hlens[b]: the reference freezes alpha beyond it.
// Phase 3: reduce 32 losses -> scalar.

#define B_    32
#define T_    1024
#define V_    1024
#define LMAX_ 128
#define S_    257
#define NEGINF (-1e30f)

__device__ __forceinline__ float lae(float a, float b) {
  // stable logaddexp, matches jnp.logaddexp (max + log1p(exp(min-max)))
  float mx = fmaxf(a, b);
  float mn = fminf(a, b);
  return mx + log1pf(expf(mn - mx));
}

// ---------------------------------------------------------------------------
// Kernel 1: norm[row] = logsumexp(ys_hat[row, 0:V]).  One wave32 per row,
// 8 x float4 per lane (32 floats), wave-local reductions via shuffles.
// ---------------------------------------------------------------------------
__global__ __launch_bounds__(256) void ctc_lse_kernel(
    const float* __restrict__ ys_hat, float* __restrict__ norm) {
  const int lane = threadIdx.x & 31;
  const int wave = threadIdx.x >> 5;
  const int row  = blockIdx.x * 8 + wave;            // [0, B*T)
  const float4* p = (const float4*)(ys_hat + (size_t)row * V_);

  float4 x[8];
#pragma unroll
  for (int i = 0; i < 8; ++i) x[i] = p[lane + 32 * i];

  float m = -INFINITY;
#pragma unroll
  for (int i = 0; i < 8; ++i)
    m = fmaxf(m, fmaxf(fmaxf(x[i].x, x[i].y), fmaxf(x[i].z, x[i].w)));
#pragma unroll
  for (int o = 16; o > 0; o >>= 1) m = fmaxf(m, __shfl_xor(m, o, 32));

  float ssum = 0.0f;
#pragma unroll
  for (int i = 0; i < 8; ++i)
    ssum += expf(x[i].x - m) + expf(x[i].y - m) +
            expf(x[i].z - m) + expf(x[i].w - m);
#pragma unroll
  for (int o = 16; o > 0; o >>= 1) ssum += __shfl_xor(ssum, o, 32);

  if (lane == 0) norm[row] = m + logf(ssum);
}

// ---------------------------------------------------------------------------
// Kernel 2: CTC forward DP. One block (288 thr = 9 wave32) per batch element;
// thread s owns extended-label state s (s < 257). Async global->LDS double-
// buffered prefetch of next timestep's gathered log-prob element per lane.
// norm[b, :] is staged in LDS so no scalar-memory wait sits in the loop.
// ---------------------------------------------------------------------------
__global__ __launch_bounds__(288) void ctc_dp_kernel(
    const float* __restrict__ ys_hat,  const int* __restrict__ ys_pad,
    const int*   __restrict__ hlens,   const int* __restrict__ ys_lens,
    const float* __restrict__ norm,    float*     __restrict__ losses) {
  __shared__ float alpha[2][288];
  __shared__ float lpbuf[2][288];
  __shared__ float norm_lds[T_];

  const int  b   = blockIdx.x;
  const int  s   = threadIdx.x;
  const bool act = (s < S_);
  const bool odd = (s & 1) != 0;

  int  lab  = 0;            // BLANK at even s (and for padding lanes)
  bool skip = false;
  if (act && odd) {
    lab = ys_pad[b * LMAX_ + (s >> 1)];
    if (s >= 3) skip = (lab != ys_pad[b * LMAX_ + (s >> 1) - 1]);
  }
  const int    hlen    = hlens[b];                   // in [T/2, T]
  const float* rowbase = ys_hat + (size_t)b * T_ * V_;
  const float* norm_b  = norm + b * T_;

  // Stage this batch element's 1024 norms into LDS.
  for (int i = s; i < T_; i += 288) norm_lds[i] = norm_b[i];

  // LDS byte offsets of this lane's private prefetch slots (flat addr low 32b).
  const uint32_t off0 = (uint32_t)(uintptr_t)&lpbuf[0][s];
  const uint32_t off1 = (uint32_t)(uintptr_t)&lpbuf[1][s];

  // t = 0 init: paths start at s=0 (blank) or s=1 (first label).
  float lp0 = rowbase[lab] - norm_b[0];
  alpha[0][s] = (act && s <= 1) ? lp0 : NEGINF;

  // Prefetch t = 1 gather element into lpbuf[1] (ASYNCcnt-tracked).
  asm volatile("global_load_async_to_lds_b32 %0, %1, off"
               :: "v"(off1), "v"(rowbase + (size_t)V_ + lab)
               : "memory");
  __syncthreads();

  int cur = 0;
  // Reference freezes alpha for t >= hlen, so those steps are no-ops: stop.
  for (int t = 1; t < hlen; ++t) {
    const int buf = t & 1;
    asm volatile("s_wait_asynccnt 0" ::: "memory");  // prefetch of buf landed
    if (t + 1 < hlen) {
      // Launch next async gather immediately; the prior read of the other
      // buffer was consumed last iteration (DScnt waited before its use), so
      // the async engine's LDS write cannot race any in-flight DS read.
      asm volatile("global_load_async_to_lds_b32 %0, %1, off"
                   :: "v"(buf ? off0 : off1),
                      "v"(rowbase + (size_t)(t + 1) * V_ + lab)
                   : "memory");
    }
    float lpv = lpbuf[buf][s];
    float nt  = norm_lds[t];
    float a    = alpha[cur][s];
    float am1  = (s >= 1) ? alpha[cur][s - 1] : NEGINF;
    float am2  = skip     ? alpha[cur][s - 2] : NEGINF;
    float cand = lae(lae(a, am1), am2) + (lpv - nt);
    alpha[cur ^ 1][s] = act ? cand : NEGINF;
    __syncthreads();
    cur ^= 1;
  }

  if (s == 0) {
    const int   L  = ys_lens[b];
    const float a1 = alpha[cur][2 * L];       // final blank
    const float a2 = alpha[cur][2 * L - 1];   // last label
    losses[b] = -lae(a1, a2);
  }
}

// ---------------------------------------------------------------------------
// Kernel 3: out = sum(losses) / B   (single wave32)
// ---------------------------------------------------------------------------
__global__ __launch_bounds__(32) void ctc_final_kernel(
    const float* __restrict__ losses, float* __restrict__ out) {
  float v = losses[threadIdx.x];
#pragma unroll
  for (int o = 16; o > 0; o >>= 1) v += __shfl_xor(v, o, 32);
  if (threadIdx.x == 0) out[0] = v * (1.0f / (float)B_);
}

extern "C" void kernel_launch(void* const* d_in, const int* in_sizes, int n_in,
                              void* d_out, int out_size, void* d_ws, size_t ws_size,
                              hipStream_t stream) {
  (void)in_sizes; (void)n_in; (void)out_size; (void)ws_size;
  const float* ys_hat  = (const float*)d_in[0];   // [B,T,V] f32
  const int*   ys_pad  = (const int*)d_in[1];     // [B,LMAX] i32
  const int*   hlens   = (const int*)d_in[2];     // [B] i32
  const int*   ys_lens = (const int*)d_in[3];     // [B] i32

  float* norm   = (float*)d_ws;                   // B*T floats
  float* losses = norm + (size_t)B_ * T_;         // B floats

  ctc_lse_kernel<<<(B_ * T_) / 8, 256, 0, stream>>>(ys_hat, norm);
  ctc_dp_kernel<<<B_, 288, 0, stream>>>(ys_hat, ys_pad, hlens, ys_lens,
                                        norm, losses);
  ctc_final_kernel<<<1, 32, 0, stream>>>(losses, (float*)d_out);
}